// MLP_t_42133629174071
// MI455X (gfx1250) — compile-verified
//
#include <hip/hip_runtime.h>

typedef float v2f __attribute__((ext_vector_type(2)));
typedef float v4f __attribute__((ext_vector_type(4)));
typedef float v8f __attribute__((ext_vector_type(8)));

#define BATCH   8192
#define F_NUM   256
#define I_NUM   128
#define O_F     128
#define B_TILE  16      // batches per workgroup; one 16x16 out tile per wave

// Fused MLP:  h[b,f] = relu(dot(X[b,f,:], W1[f,:]) + b1[f])
//             out[b,o] = relu(sum_f h[b,f]*W2[o,f] + b2[o])
//
// Memory-bound (1 GiB input @ 23.3 TB/s => ~46us floor). Stage 1 streams the
// input with wave-per-row b128 loads + VALU dot + wave32 shfl reduce; stage 2
// is a WMMA f32 16x16x4 GEMM entirely out of LDS.
__global__ __launch_bounds__(256) void mlp_fused_kernel(
    const float* __restrict__ inputs,   // [B, F, I]
    const float* __restrict__ W1,       // [F, I]
    const float* __restrict__ b1,       // [F]
    const float* __restrict__ W2,       // [O, F]
    const float* __restrict__ b2,       // [O]
    float* __restrict__ out)            // [B, O]
{
    // hT[f][b_local]: transposed so stage-2 A-fragment reads are 32 distinct banks
    __shared__ float hT[F_NUM * B_TILE];      // 16 KB
    // w2t[k][o] = W2[o][k]: stage-2 B-fragment reads hit 16 distinct banks/half-wave
    __shared__ float w2t[F_NUM * O_F];        // 128 KB

    const int tid   = threadIdx.x;
    const int wave  = tid >> 5;               // 0..7 (wave32)
    const int lane  = tid & 31;
    const int bBase = blockIdx.x * B_TILE;

    // ---- Stage W2 -> LDS (transposed). 128 KB, L2-resident across all WGs. ----
    for (int idx = tid; idx < F_NUM * O_F; idx += 256) {
        const int o = idx & (O_F - 1);
        const int k = idx >> 7;               // idx = k*O_F + o  (coalesced LDS writes)
        w2t[idx] = W2[o * F_NUM + k];
    }

    // ---- Stage 1: one (b,f) row per wave-iteration.
    // Wave w owns b_local {2w, 2w+1}; f runs innermost so each wave streams two
    // contiguous 128 KB input regions. Lane l covers i = 4l..4l+3 (512 B/row).
    #pragma unroll 4
    for (int j = 0; j < 2 * F_NUM; ++j) {
        const int bl = (wave << 1) + (j >> 8);
        const int f  = j & (F_NUM - 1);
        const float* rowp = inputs + ((size_t)(bBase + bl) * F_NUM + f) * I_NUM + lane * 4;
        const v4f x = *(const v4f*)rowp;                       // global_load_b128
        const v4f w = *(const v4f*)(W1 + f * I_NUM + lane * 4);// L2/WGP$-hot
        float p = x.x * w.x + x.y * w.y + x.z * w.z + x.w * w.w;
        // wave32 tree reduction
        p += __shfl_xor(p, 16, 32);
        p += __shfl_xor(p,  8, 32);
        p += __shfl_xor(p,  4, 32);
        p += __shfl_xor(p,  2, 32);
        p += __shfl_xor(p,  1, 32);
        if (lane == 0) {
            const float h = p + b1[f];
            hT[f * B_TILE + bl] = h > 0.0f ? h : 0.0f;         // relu
        }
    }

    __syncthreads();

    // ---- Stage 2: wave w computes out tile [bBase..+15] x [16w..16w+15]
    // via 64 chained V_WMMA_F32_16X16X4_F32 (K = 256).
    const int nB   = wave * 16;
    const int half = lane >> 4;     // 0: K={k0,k0+1}; 1: K={k0+2,k0+3}
    const int lm   = lane & 15;     // M index for A, N index for B/C/D

    v8f c = {};
    #pragma unroll 8
    for (int k0 = 0; k0 < F_NUM; k0 += 4) {
        const int ka = k0 + 2 * half;
        v2f a, b;
        a.x = hT[ ka      * B_TILE + lm];       // A[m=lm, k=ka]
        a.y = hT[(ka + 1) * B_TILE + lm];       // A[m=lm, k=ka+1]
        b.x = w2t[ ka      * O_F + nB + lm];    // B[k=ka,   n=lm] = W2[nB+lm, ka]
        b.y = w2t[(ka + 1) * O_F + nB + lm];    // B[k=ka+1, n=lm]
        // 8 args: (neg_a, A, neg_b, B, c_mod, C, reuse_a, reuse_b)
        c = __builtin_amdgcn_wmma_f32_16x16x4_f32(
                false, a, false, b, (short)0, c, false, false);
    }

    const float bias = b2[nB + lm];
    #pragma unroll
    for (int g = 0; g < 8; ++g) {
        const int m = g + 8 * half;             // D VGPR layout (ISA 7.12.2)
        float v = c[g] + bias;
        v = v > 0.0f ? v : 0.0f;                // relu
        out[(size_t)(bBase + m) * O_F + nB + lm] = v;
    }
}

extern "C" void kernel_launch(void* const* d_in, const int* in_sizes, int n_in,
                              void* d_out, int out_size, void* d_ws, size_t ws_size,
                              hipStream_t stream) {
    const float* inputs = (const float*)d_in[0];
    const float* W1     = (const float*)d_in[1];
    const float* b1     = (const float*)d_in[2];
    const float* W2     = (const float*)d_in[3];
    const float* b2     = (const float*)d_in[4];
    float* out          = (float*)d_out;

    dim3 grid(BATCH / B_TILE);   // 512 workgroups
    dim3 block(256);             // 8 wave32s
    mlp_fused_kernel<<<grid, block, 0, stream>>>(inputs, W1, b1, W2, b2, out);
}